// KineticForecastingFramework_30236569764433
// MI455X (gfx1250) — compile-verified
//
#include <hip/hip_runtime.h>
#include <stdint.h>

#define Q 32
#define DT_C 0.1f
#define XI_STEP (75.0f / 31.0f)

#define EPW 128                 // edges per wave
#define WPB 8                   // waves per block (256 threads)
#define EPB (EPW * WPB)         // 1024 edges per block

// ---------------------------------------------------------------------------
// Guaranteed-hardware no-return fp32 atomic add: saddr (SGPR pair) + 32-bit
// VGPR byte offset. Tracked by STOREcnt; s_endpgm's implicit wait-idle covers
// completion, and nothing in-kernel reads acc, so no explicit wait needed.
// ---------------------------------------------------------------------------
__device__ __forceinline__ void gatomic_fadd(unsigned long long sbase,
                                             uint32_t voff, float v) {
  asm volatile("global_atomic_add_f32 %0, %1, %2"
               :: "v"(voff), "v"(v), "s"(sbase) : "memory");
}

// ---------------------------------------------------------------------------
// K3 (hot kernel, listed first): edge transport accumulation.
//   One wave per edge, lane = channel q (wave32 == Q -> perfect 128B rows).
//   Per-wave 128-edge metadata tile staged into LDS via CDNA5 async-to-LDS
//   copies (ASYNCcnt / s_wait_asynccnt), packed 16B/edge so read-back is a
//   single ds_load_b128. Metadata is then broadcast to SGPRs with
//   v_readfirstlane so all address math runs on the SALU and the invdeg
//   lookup becomes an SMEM s_load. All state is L2-resident (f 12.8MB,
//   acc 12.8MB, invdeg 0.4MB << 192MB L2).
//   acc[src] += c*(f[dst]-f[src]) ; acc[dst] -= same  (xi scale deferred)
// ---------------------------------------------------------------------------
__global__ __launch_bounds__(256) void k_edges(
    const float* __restrict__ f,
    const float* __restrict__ w,
    const int* __restrict__ src,
    const int* __restrict__ dst,
    const float* __restrict__ invdeg,
    float* __restrict__ acc,
    int E)
{
  __shared__ int4 s_meta[EPB];   // .x=src .y=dst .z=w(bits) .w=pad

  const int tid  = threadIdx.x;
  const int lane = tid & 31;
  const int wv   = __builtin_amdgcn_readfirstlane(tid >> 5);   // force uniform
  const long long ebase = (long long)blockIdx.x * EPB + (long long)wv * EPW;

  // LDS byte offset of this wave's slice (low 32 bits of generic LDS address)
  const uint32_t l_meta = (uint32_t)(size_t)(void*)(&s_meta[wv * EPW]);

  const unsigned long long b_src = (unsigned long long)(size_t)src;
  const unsigned long long b_dst = (unsigned long long)(size_t)dst;
  const unsigned long long b_w   = (unsigned long long)(size_t)w;

  // Stage 128 edges: 4 chunks x 32 lanes, 3 async b32 copies per lane into an
  // interleaved 16B/edge LDS layout. Wave reads only its own slice, so only
  // s_wait_asynccnt is required (no workgroup barrier).
  #pragma unroll
  for (int c = 0; c < 4; ++c) {
    const int j = c * 32 + lane;
    long long e = ebase + j;
    if (e >= E) e = (long long)E - 1;          // tail clamp (harmless dup)
    const uint32_t gb = (uint32_t)(e << 2);    // global byte offset
    const uint32_t lo = l_meta + (uint32_t)(j * 16);
    asm volatile("global_load_async_to_lds_b32 %0, %1, %2"
                 :: "v"(lo),     "v"(gb), "s"(b_src) : "memory");
    asm volatile("global_load_async_to_lds_b32 %0, %1, %2"
                 :: "v"(lo + 4), "v"(gb), "s"(b_dst) : "memory");
    asm volatile("global_load_async_to_lds_b32 %0, %1, %2"
                 :: "v"(lo + 8), "v"(gb), "s"(b_w)   : "memory");
  }
  asm volatile("s_wait_asynccnt 0x0" ::: "memory");

  const uint32_t lane4 = (uint32_t)(lane << 2);        // loop-invariant voffset
  const int base = wv * EPW;

  #pragma unroll 4
  for (int j = 0; j < EPW; ++j) {
    if (ebase + j >= E) break;
    const int4 md = s_meta[base + j];                  // one ds_load_b128
    const int   su = __builtin_amdgcn_readfirstlane(md.x);
    const int   du = __builtin_amdgcn_readfirstlane(md.y);
    const float wu = __builtin_bit_cast(float, __builtin_amdgcn_readfirstlane(md.z));

    const float c  = wu * invdeg[su];                  // SMEM s_load (uniform)
    const float fs = fmaxf(f[(size_t)su * Q + lane], 0.0f);  // saddr+lane form
    const float fd = fmaxf(f[(size_t)du * Q + lane], 0.0f);
    const float m  = c * (fd - fs);

    gatomic_fadd((unsigned long long)(size_t)(acc + (size_t)su * Q), lane4,  m);
    gatomic_fadd((unsigned long long)(size_t)(acc + (size_t)du * Q), lane4, -m);
  }
}

// ---------------------------------------------------------------------------
// K0: zero scratch, 128-bit stores (plus scalar tail kernel)
// ---------------------------------------------------------------------------
__global__ void k_zero4(uint4* __restrict__ p, int n4) {
  int i = blockIdx.x * blockDim.x + threadIdx.x;
  if (i < n4) p[i] = make_uint4(0u, 0u, 0u, 0u);
}
__global__ void k_zero1(uint32_t* __restrict__ p, int n) {
  int i = blockIdx.x * blockDim.x + threadIdx.x;
  if (i < n) p[i] = 0u;
}

// ---------------------------------------------------------------------------
// K1: out-degree histogram (integer atomics -> always hardware)
// ---------------------------------------------------------------------------
__global__ void k_degree(const int* __restrict__ src, uint32_t* __restrict__ deg, int E) {
  int e = blockIdx.x * blockDim.x + threadIdx.x;
  if (e < E) atomicAdd(&deg[src[e]], 1u);
}

// ---------------------------------------------------------------------------
// K2: invdeg[i] = 1 / max(deg[i], 1)
// ---------------------------------------------------------------------------
__global__ void k_invdeg(const uint32_t* __restrict__ deg, float* __restrict__ invdeg, int n) {
  int i = blockIdx.x * blockDim.x + threadIdx.x;
  if (i < n) invdeg[i] = 1.0f / fmaxf((float)deg[i], 1.0f);
}

// ---------------------------------------------------------------------------
// K4: out = relu( relu(f) - DT*(xi[q]*acc - coll - srcterm) ), float4 streams.
//   flat index 4i..4i+3 -> channels q = 4*(i&7)+k  (Q=32, row-major [N,Q])
// ---------------------------------------------------------------------------
__global__ void k_final4(const float4* __restrict__ fdist,
                         const float4* __restrict__ coll,
                         const float4* __restrict__ srct,
                         const float4* __restrict__ acc,
                         float4* __restrict__ out, int n4) {
  int i = blockIdx.x * blockDim.x + threadIdx.x;
  if (i < n4) {
    const float q0 = (float)((i & 7) << 2) * XI_STEP;
    const float4 a = acc[i], fv = fdist[i], cl = coll[i], st = srct[i];
    float4 r;
    r.x = fmaxf(fmaxf(fv.x, 0.0f) - DT_C * ((q0 + 0.0f * XI_STEP) * a.x - cl.x - st.x), 0.0f);
    r.y = fmaxf(fmaxf(fv.y, 0.0f) - DT_C * ((q0 + 1.0f * XI_STEP) * a.y - cl.y - st.y), 0.0f);
    r.z = fmaxf(fmaxf(fv.z, 0.0f) - DT_C * ((q0 + 2.0f * XI_STEP) * a.z - cl.z - st.z), 0.0f);
    r.w = fmaxf(fmaxf(fv.w, 0.0f) - DT_C * ((q0 + 3.0f * XI_STEP) * a.w - cl.w - st.w), 0.0f);
    out[i] = r;
  }
}

// ---------------------------------------------------------------------------
// Launch. Inputs (setup_inputs order):
//   0 f_distribution [N*Q] f32, 1 collision [N*Q] f32, 2 source [N*Q] f32,
//   3 edge_weight [E] f32, 4 src [E] i32, 5 dst [E] i32.
// Scratch: deg (N u32) | invdeg (N f32) | acc (N*Q f32), re-zeroed each call.
// ---------------------------------------------------------------------------
extern "C" void kernel_launch(void* const* d_in, const int* in_sizes, int n_in,
                              void* d_out, int out_size, void* d_ws, size_t ws_size,
                              hipStream_t stream) {
  (void)n_in; (void)out_size; (void)ws_size;

  const float* fdist = (const float*)d_in[0];
  const float* coll  = (const float*)d_in[1];
  const float* srct  = (const float*)d_in[2];
  const float* ew    = (const float*)d_in[3];
  const int*   src   = (const int*)d_in[4];
  const int*   dst   = (const int*)d_in[5];

  const int NQ = in_sizes[0];
  const int N  = NQ / Q;
  const int E  = in_sizes[3];

  uint32_t* deg    = (uint32_t*)d_ws;
  float*    invdeg = (float*)(deg + N);
  float*    acc    = invdeg + N;

  // zero deg + invdeg slot + acc every call (deterministic across replays)
  const int zn  = 2 * N + NQ;
  const int zn4 = zn >> 2, zr = zn & 3;
  k_zero4<<<(zn4 + 255) / 256, 256, 0, stream>>>((uint4*)d_ws, zn4);
  if (zr) k_zero1<<<1, 256, 0, stream>>>((uint32_t*)d_ws + 4 * zn4, zr);

  k_degree<<<(E + 255) / 256, 256, 0, stream>>>(src, deg, E);
  k_invdeg<<<(N + 255) / 256, 256, 0, stream>>>(deg, invdeg, N);

  const int nblk = (E + EPB - 1) / EPB;
  k_edges<<<nblk, 256, 0, stream>>>(fdist, ew, src, dst, invdeg, acc, E);

  const int n4 = NQ >> 2;
  k_final4<<<(n4 + 255) / 256, 256, 0, stream>>>(
      (const float4*)fdist, (const float4*)coll, (const float4*)srct,
      (const float4*)acc, (float4*)d_out, n4);
}